// ImprovedGCN_19026705121711
// MI455X (gfx1250) — compile-verified
//
#include <hip/hip_runtime.h>

typedef __attribute__((ext_vector_type(2))) float v2f;
typedef __attribute__((ext_vector_type(8))) float v8f;

#define BN_EPS 1e-5f

// ---------------- degree / normalization ----------------

__global__ void fill_ones(float* p, int n) {
  int i = blockIdx.x * blockDim.x + threadIdx.x;
  if (i < n) p[i] = 1.0f;
}

__global__ void deg_accum(const int* __restrict__ dst, float* deg, int E) {
  int i = blockIdx.x * blockDim.x + threadIdx.x;
  if (i < E) atomicAdd(&deg[dst[i]], 1.0f);
}

__global__ void deg_to_dinv(float* deg, int n) {
  int i = blockIdx.x * blockDim.x + threadIdx.x;
  if (i < n) deg[i] = rsqrtf(deg[i]);
}

// ---------------- WMMA f32 GEMM: out[N,Fout] = H[N,Fin] @ W[Fin,Fout] ----------------
// Block = 256 threads = 8 waves. Each wave owns a 16x16 output tile; block covers
// 128 rows x 16 cols. The Fin x 16 slice of W is staged in LDS (<= 8KB) and reused
// by all 8 waves. K-loop of V_WMMA_F32_16X16X4_F32 (full f32 precision).
__global__ void __launch_bounds__(256) gemm_wmma_f32(
    const float* __restrict__ H, const float* __restrict__ W,
    float* __restrict__ out, int Nrows, int Fin, int Fout) {
  __shared__ float Wl[128 * 16];

  const int tid  = threadIdx.x;
  const int col0 = blockIdx.x << 4;

  // cooperative stage of W tile [Fin x 16] into LDS (all waves participate)
  for (int i = tid; i < (Fin << 4); i += 256) {
    int kk = i >> 4, cc = i & 15;
    Wl[i] = W[kk * Fout + col0 + cc];
  }
  __syncthreads();

  const int wave = tid >> 5;
  const int lane = tid & 31;
  const int row0 = ((blockIdx.y << 3) + wave) << 4;
  if (row0 + 16 > Nrows) return;   // wave-uniform exit, after the only barrier

  const int half = lane >> 4;      // 0: K=0,1  1: K=2,3 (16x4 f32 A layout)
  const int l    = lane & 15;

  const float* arow = H + (size_t)(row0 + l) * Fin + (half << 1);
  v8f c = {};
  for (int k = 0; k < Fin; k += 4) {
    v2f a = *(const v2f*)(arow + k);       // 2 consecutive f32 of row M=row0+l
    int kb = k + (half << 1);
    v2f b;
    b.x = Wl[(kb + 0) * 16 + l];           // B 4x16: VGPR0 K=0/2, VGPR1 K=1/3
    b.y = Wl[(kb + 1) * 16 + l];
    c = __builtin_amdgcn_wmma_f32_16x16x4_f32(false, a, false, b,
                                              (short)0, c, false, false);
  }

  // C/D layout: VGPR i -> row M = i + 8*half, col = lane&15
  float* orow = out + (size_t)(row0 + (half << 3)) * Fout + col0 + l;
#pragma unroll
  for (int i = 0; i < 8; i++) orow[(size_t)i * Fout] = c[i];
}

// ---------------- aggregation ----------------

// out[n,f] = dinv[n]^2 * hw[n,f] + bias[f]   (self-loop term + conv bias; no zero-init pass)
__global__ void selfloop_bias(const float* __restrict__ hw, const float* __restrict__ dinv,
                              const float* __restrict__ bias, float* __restrict__ out,
                              long long total, int logF) {
  long long i = (long long)blockIdx.x * blockDim.x + threadIdx.x;
  if (i >= total) return;
  int f = (int)(i & ((1 << logF) - 1));
  int n = (int)(i >> logF);
  float di = dinv[n];
  out[i] = di * di * hw[i] + bias[f];
}

// out[dst,f] += dinv[src]*dinv[dst] * hw[src,f]   one thread per (edge, feature)
__global__ void agg_edges(const float* __restrict__ hw, const int* __restrict__ src,
                          const int* __restrict__ dst, const float* __restrict__ dinv,
                          float* __restrict__ out, long long total, int logF) {
  long long i = (long long)blockIdx.x * blockDim.x + threadIdx.x;
  if (i >= total) return;
  int f = (int)(i & ((1 << logF) - 1));
  int e = (int)(i >> logF);
  int s = src[e], d = dst[e];
  float w = dinv[s] * dinv[d];
  atomicAdd(out + (((long long)d << logF) + f), w * hw[((long long)s << logF) + f]);
}

// ---------------- batchnorm (training-mode batch stats) + relu ----------------

__global__ void zero_f(float* p, int n) {
  int i = blockIdx.x * blockDim.x + threadIdx.x;
  if (i < n) p[i] = 0.0f;
}

// per-feature sum / sum-of-squares: thread owns feature f = tid & (F-1), strides rows
__global__ void bn_stats(const float* __restrict__ h, float* sums, int N, int logF) {
  int tid = blockIdx.x * blockDim.x + threadIdx.x;
  int F = 1 << logF;
  int f = tid & (F - 1);
  int r0 = tid >> logF;
  int rstride = (gridDim.x * blockDim.x) >> logF;
  float s = 0.0f, s2 = 0.0f;
  for (int r = r0; r < N; r += rstride) {
    float x = h[((long long)r << logF) + f];
    s += x; s2 += x * x;
  }
  atomicAdd(&sums[f], s);
  atomicAdd(&sums[F + f], s2);
}

__global__ void bn_apply_relu(float* h, const float* __restrict__ sums,
                              const float* __restrict__ g, const float* __restrict__ be,
                              long long total, int logF, float invN) {
  long long i = (long long)blockIdx.x * blockDim.x + threadIdx.x;
  if (i >= total) return;
  int F = 1 << logF;
  int f = (int)(i & (F - 1));
  float m = sums[f] * invN;
  float v = sums[F + f] * invN - m * m;
  float sc = g[f] * rsqrtf(v + BN_EPS);
  float y = (h[i] - m) * sc + be[f];
  h[i] = fmaxf(y, 0.0f);
}

// ---------------- final FC: [N,32] @ [32,1] + b ----------------

__global__ void fc_out(const float* __restrict__ h, const float* __restrict__ Wfc,
                       const float* __restrict__ bfc, float* __restrict__ out, int N) {
  int n = blockIdx.x * blockDim.x + threadIdx.x;
  if (n >= N) return;
  const float4* hp = (const float4*)(h + (size_t)n * 32);
  const float4* wp = (const float4*)Wfc;
  float acc = bfc[0];
#pragma unroll
  for (int i = 0; i < 8; i++) {
    float4 a = hp[i]; float4 w = wp[i];
    acc += a.x * w.x + a.y * w.y + a.z * w.z + a.w * w.w;
  }
  out[n] = acc;
}

// ---------------- launch ----------------

extern "C" void kernel_launch(void* const* d_in, const int* in_sizes, int n_in,
                              void* d_out, int out_size, void* d_ws, size_t ws_size,
                              hipStream_t stream) {
  const float* x   = (const float*)d_in[0];
  const int* edge  = (const int*)d_in[1];
  const float* W1  = (const float*)d_in[2];
  const float* b1  = (const float*)d_in[3];
  const float* g1  = (const float*)d_in[4];
  const float* be1 = (const float*)d_in[5];
  const float* W2  = (const float*)d_in[6];
  const float* b2  = (const float*)d_in[7];
  const float* g2  = (const float*)d_in[8];
  const float* be2 = (const float*)d_in[9];
  const float* W3  = (const float*)d_in[10];
  const float* b3  = (const float*)d_in[11];
  const float* g3  = (const float*)d_in[12];
  const float* be3 = (const float*)d_in[13];
  const float* Wfc = (const float*)d_in[14];
  const float* bfc = (const float*)d_in[15];
  float* outp = (float*)d_out;

  const int N = in_sizes[0] / 128;
  const int E = in_sizes[1] / 2;
  const int* src = edge;
  const int* dst = edge + E;

  // workspace layout (floats): dinv[N] | sums[256] | bufA[N*128] | bufB[N*128]
  size_t npad = ((size_t)N + 127) & ~(size_t)127;
  float* dinv = (float*)d_ws;
  float* sums = dinv + npad;
  float* bufA = sums + 256;
  float* bufB = bufA + (size_t)N * 128;

  const int TB = 256;
  int gN = (N + TB - 1) / TB;
  int gE = (E + TB - 1) / TB;

  // symmetric GCN normalization: deg = 1 + indeg; dinv = deg^-1/2
  fill_ones<<<gN, TB, 0, stream>>>(dinv, N);
  deg_accum<<<gE, TB, 0, stream>>>(dst, dinv, E);
  deg_to_dinv<<<gN, TB, 0, stream>>>(dinv, N);

  auto layer = [&](const float* hin, int Fin, int Fout, int logF,
                   const float* W, const float* b, const float* g, const float* be) {
    dim3 gg(Fout / 16, (unsigned)((N / 16 + 7) / 8));
    gemm_wmma_f32<<<gg, 256, 0, stream>>>(hin, W, bufA, N, Fin, Fout);

    long long totN = (long long)N << logF;
    int gbN = (int)((totN + TB - 1) / TB);
    selfloop_bias<<<gbN, TB, 0, stream>>>(bufA, dinv, b, bufB, totN, logF);

    long long totE = (long long)E << logF;
    int gbE = (int)((totE + TB - 1) / TB);
    agg_edges<<<gbE, TB, 0, stream>>>(bufA, src, dst, dinv, bufB, totE, logF);

    zero_f<<<1, TB, 0, stream>>>(sums, 2 << logF);
    bn_stats<<<512, TB, 0, stream>>>(bufB, sums, N, logF);
    bn_apply_relu<<<gbN, TB, 0, stream>>>(bufB, sums, g, be, totN, logF, 1.0f / (float)N);
  };

  layer(x,    128, 128, 7, W1, b1, g1, be1);   // bufB = h1 [N,128]
  layer(bufB, 128,  64, 6, W2, b2, g2, be2);   // bufB = h2 [N,64]
  layer(bufB,  64,  32, 5, W3, b3, g3, be3);   // bufB = h3 [N,32]

  fc_out<<<gN, TB, 0, stream>>>(bufB, Wfc, bfc, outp, N);
}